// CompGraphConv_23106924053360
// MI455X (gfx1250) — compile-verified
//
#include <hip/hip_runtime.h>
#include <cstdint>
#include <cstddef>

// ======================= CDNA5 fp32 WMMA helpers =======================
typedef float v2f __attribute__((ext_vector_type(2)));
typedef float v8f __attribute__((ext_vector_type(8)));

#define DF 200                    // feature dim
#define KSTEPS (DF / 4)           // 50 k-steps of V_WMMA_F32_16X16X4_F32
#define CTILES ((DF + 15) / 16)   // 13 column tiles (last one masked)
#define EPB 64                    // edges per block in edge-att kernel
#define KPAD 202                  // padded LDS row stride (conflict-free A reads)

static __device__ __forceinline__ v8f wmma_f32(v2f a, v2f b, v8f c) {
  // D = A(16x4) * B(4x16) + C ; args: neg_a, A, neg_b, B, c_mod, C, reuse_a, reuse_b
  return __builtin_amdgcn_wmma_f32_16x16x4_f32(false, a, false, b, (short)0, c,
                                               false, false);
}

// ======================= small prep kernels =======================

// M[r] = sum_b w_comp[r,b] * relation_att[b]   (R x D x D, resident in 192MB L2)
__global__ __launch_bounds__(256) void relM_kernel(
    const float* __restrict__ relA, const float* __restrict__ wcomp,
    float* __restrict__ Mrel, int B) {
  const int r = blockIdx.y;
  const int idx = blockIdx.x * 256 + threadIdx.x;
  if (idx >= DF * DF) return;
  float acc = 0.f;
  for (int b = 0; b < B; ++b)
    acc = fmaf(wcomp[r * B + b], relA[(size_t)b * DF * DF + idx], acc);
  Mrel[(size_t)r * DF * DF + idx] = acc;
}

// sbias[n] = wS_b[n] - sum_k loop_rel[k] * wS_w[n,k]  (folds self-loop subtract)
__global__ __launch_bounds__(256) void sbias_kernel(
    const float* __restrict__ loop, const float* __restrict__ wSw,
    const float* __restrict__ wSb, float* __restrict__ sbias) {
  const int n = threadIdx.x;
  if (n >= DF) return;
  float acc = 0.f;
  const float* w = wSw + (size_t)n * DF;
  for (int k = 0; k < DF; ++k) acc = fmaf(loop[k], w[k], acc);
  sbias[n] = wSb[n] - acc;
}

// ======================= node projections (WMMA + LDS B panels) ==========
// Block = 8 waves covering 8 row tiles, all sharing ONE column tile.
// The 4 weight panels (k/q/v/s, 16 cols x 200 k) are staged once into LDS.
__global__ __launch_bounds__(256) void node_proj_wmma(
    const float* __restrict__ X,
    const float* __restrict__ kw, const float* __restrict__ kb,
    const float* __restrict__ qw, const float* __restrict__ qb,
    const float* __restrict__ vw, const float* __restrict__ vb,
    const float* __restrict__ sw, const float* __restrict__ sb2,
    float* __restrict__ Kf, float* __restrict__ Qf,
    float* __restrict__ Vf, float* __restrict__ Sf, int nrows) {
  __shared__ v2f Bsh[4 * KSTEPS * 2 * 16];  // 51,200 B
  const int t = threadIdx.x;
  const int ct = blockIdx.y;

  // stage B panels: Bsh[((mat*50+kk)*2+half)*16 + c] = {W[col, k0], W[col, k0+1]}
  auto stage = [&](const float* __restrict__ W, int mat) {
    for (int idx = t; idx < 16 * (DF / 2); idx += 256) {
      const int c = idx / (DF / 2), kp = idx % (DF / 2);
      const int col = min(ct * 16 + c, DF - 1);  // clamped; garbage cols masked
      const v2f v = *(const v2f*)&W[(size_t)col * DF + 2 * kp];
      Bsh[((mat * KSTEPS + (kp >> 1)) * 2 + (kp & 1)) * 16 + c] = v;
    }
  };
  stage(kw, 0); stage(qw, 1); stage(vw, 2); stage(sw, 3);
  __syncthreads();

  const int lane = t & 31, wid = t >> 5;
  const int row_tiles = (nrows + 15) >> 4;
  const int rt = blockIdx.x * 8 + wid;
  if (rt >= row_tiles) return;
  const int half = lane >> 4, l16 = lane & 15;
  const int arow = min((rt << 4) + l16, nrows - 1);
  const float* xr = X + (size_t)arow * DF;

  v8f ck = {}, cq = {}, cv = {}, cs = {};
  for (int kk = 0; kk < KSTEPS; ++kk) {
    const int k0 = (kk << 2) + (half << 1);
    v2f a = *(const v2f*)&xr[k0];
    const int bi = (kk * 2 + half) * 16 + l16;
    ck = wmma_f32(a, Bsh[(0 * KSTEPS * 2) * 16 + bi], ck);
    cq = wmma_f32(a, Bsh[(1 * KSTEPS * 2) * 16 + bi], cq);
    cv = wmma_f32(a, Bsh[(2 * KSTEPS * 2) * 16 + bi], cv);
    cs = wmma_f32(a, Bsh[(3 * KSTEPS * 2) * 16 + bi], cs);
  }

  const int col = (ct << 4) + l16;
  if (col < DF) {
    const float bkb = kb[col], bqb = qb[col], bvb = vb[col], bsb = sb2[col];
#pragma unroll
    for (int j = 0; j < 8; ++j) {
      const int row = (rt << 4) + j + (half << 3);
      if (row < nrows) {
        const size_t off = (size_t)row * DF + col;
        Kf[off] = ck[j] + bkb;
        Qf[off] = cq[j] + bqb;
        Vf[off] = cv[j] + bvb;
        Sf[off] = cs[j] + bsb;
      }
    }
  }
}

// Generic Y = X @ W^T + bias (for r_out = r_feats @ wR^T + wR_b; small)
__global__ __launch_bounds__(256) void gemm_xwT_wmma(
    const float* __restrict__ X, const float* __restrict__ W,
    const float* __restrict__ bias, float* __restrict__ Y, int mrows) {
  const int lane = threadIdx.x & 31;
  const int wid = threadIdx.x >> 5;
  const int tile = blockIdx.x * 8 + wid;
  const int row_tiles = (mrows + 15) >> 4;
  if (tile >= row_tiles * CTILES) return;
  const int rt = tile / CTILES, ct = tile % CTILES;
  const int half = lane >> 4, l16 = lane & 15;
  const int arow = min((rt << 4) + l16, mrows - 1);
  const int bcol = min((ct << 4) + l16, DF - 1);
  const float* xr = X + (size_t)arow * DF;
  const float* wr = W + (size_t)bcol * DF;
  v8f c = {};
  for (int kk = 0; kk < KSTEPS; ++kk) {
    const int k0 = (kk << 2) + (half << 1);
    v2f a = *(const v2f*)&xr[k0];
    v2f b = *(const v2f*)&wr[k0];
    c = wmma_f32(a, b, c);
  }
  const int col = (ct << 4) + l16;
  if (col < DF) {
    const float bb = bias[col];
#pragma unroll
    for (int j = 0; j < 8; ++j) {
      const int row = (rt << 4) + j + (half << 3);
      if (row < mrows) Y[(size_t)row * DF + col] = c[j] + bb;
    }
  }
}

// ======================= edge bucketing by relation type ==================
__global__ __launch_bounds__(256) void init_kernel(
    float* __restrict__ agg, float* __restrict__ mseg,
    float* __restrict__ denom, float* __restrict__ colsum,
    float* __restrict__ colsq, int* __restrict__ cnt, int R, int N,
    size_t total) {
  const size_t i = (size_t)blockIdx.x * 256 + threadIdx.x;
  if (i < total) agg[i] = 0.f;
  if (i < (size_t)N) { mseg[i] = -3.0e38f; denom[i] = 0.f; }
  if (i < 256) { colsum[i] = 0.f; colsq[i] = 0.f; }
  if (i < (size_t)R) cnt[i] = 0;
}

__global__ __launch_bounds__(256) void hist_kernel(const int* __restrict__ ety,
                                                   int* __restrict__ cnt,
                                                   int E) {
  const int e = blockIdx.x * 256 + threadIdx.x;
  if (e < E) atomicAdd(&cnt[ety[e]], 1);
}

__global__ __launch_bounds__(256) void scan_kernel(const int* __restrict__ cnt,
                                                   int* __restrict__ offs,
                                                   int* __restrict__ cursor,
                                                   int R) {
  __shared__ int sh[512];
  const int t = threadIdx.x;
  if (t < R) sh[t] = cnt[t];
  __syncthreads();
  if (t == 0) {  // R ~ 100: trivial serial exclusive scan
    int acc = 0;
    for (int i = 0; i < R; ++i) { const int c = sh[i]; sh[i] = acc; acc += c; }
    sh[R] = acc;
  }
  __syncthreads();
  if (t <= R) offs[t] = sh[t];
  if (t < R) cursor[t] = sh[t];
}

__global__ __launch_bounds__(256) void scatter_kernel(
    const int* __restrict__ ety, int* __restrict__ cursor,
    int* __restrict__ eidx, int E) {
  const int e = blockIdx.x * 256 + threadIdx.x;
  if (e < E) {
    const int pos = atomicAdd(&cursor[ety[e]], 1);
    eidx[pos] = e;
  }
}

// ======================= edge bilinear attention (WMMA) ==================
// Block: 4 waves, 64 same-type edges (4 groups of 16). K rows staged in LDS,
// M_t column panels staged per 16-col tile (read ONCE per 64 edges).
// U = K_g @ M_t via v_wmma_f32_16x16x4_f32 ; att = rowsum(U * Q_g).
__global__ __launch_bounds__(128) void edge_att_wmma(
    const float* __restrict__ Kf, const float* __restrict__ Qf,
    const float* __restrict__ Mrel, const int* __restrict__ eidx,
    const int* __restrict__ offs, float* __restrict__ att) {
  __shared__ float Ksh[EPB * KPAD];  // 51,712 B
  __shared__ float Bsh[DF * 16];     // 12,800 B
  __shared__ int esh[EPB];
  const int r = blockIdx.x;
  const int start = offs[r], end = offs[r + 1];
  const int estart = start + blockIdx.y * EPB;
  if (estart >= end) return;
  const int count = min(EPB, end - estart);
  const int t = threadIdx.x;

  if (t < EPB) esh[t] = (t < count) ? eidx[estart + t] : -1;
  __syncthreads();
  // gather K rows (padded rows are zero -> zero U rows, stores masked)
  for (int idx = t; idx < EPB * DF; idx += 128) {
    const int m = idx / DF, k = idx % DF;
    const int e = esh[m];
    Ksh[m * KPAD + k] = (e >= 0) ? Kf[(size_t)e * DF + k] : 0.f;
  }

  const int lane = t & 31, wid = t >> 5;  // wid = edge-group 0..3
  const int half = lane >> 4, l16 = lane & 15;
  const float* Mr = Mrel + (size_t)r * DF * DF;
  const float* arow = &Ksh[(wid * 16 + l16) * KPAD];
  float sacc[8] = {0.f, 0.f, 0.f, 0.f, 0.f, 0.f, 0.f, 0.f};

  for (int ctile = 0; ctile < CTILES; ++ctile) {
    __syncthreads();  // Ksh ready (1st iter) / previous Bsh reads done
    for (int idx = t; idx < DF * 16; idx += 128) {
      const int k = idx >> 4, c = idx & 15;
      const int col = min(ctile * 16 + c, DF - 1);
      Bsh[idx] = Mr[(size_t)k * DF + col];  // coalesced rows of M_t
    }
    __syncthreads();

    v8f u = {};
    for (int kk = 0; kk < KSTEPS; ++kk) {
      const int k0 = (kk << 2) + (half << 1);
      v2f a = *(const v2f*)&arow[k0];
      v2f b;
      b.x = Bsh[k0 * 16 + l16];
      b.y = Bsh[(k0 + 1) * 16 + l16];
      u = wmma_f32(a, b, u);
    }
    const int col = ctile * 16 + l16;
    if (col < DF) {
#pragma unroll
      for (int j = 0; j < 8; ++j) {
        const int m = wid * 16 + j + (half << 3);
        const int e = esh[m];
        const float qv = (e >= 0) ? Qf[(size_t)e * DF + col] : 0.f;
        sacc[j] = fmaf(u[j], qv, sacc[j]);
      }
    }
  }
  // reduce each row's partial dot across the 16 lanes of its half
#pragma unroll
  for (int j = 0; j < 8; ++j) {
    float v = sacc[j];
    v += __shfl_xor(v, 8, 16);
    v += __shfl_xor(v, 4, 16);
    v += __shfl_xor(v, 2, 16);
    v += __shfl_xor(v, 1, 16);
    if (l16 == 0) {
      const int e = esh[wid * 16 + j + (half << 3)];
      if (e >= 0) att[e] = v;
    }
  }
}

// ======================= segment softmax + aggregation ==================
__global__ __launch_bounds__(256) void attmax_kernel(
    const float* __restrict__ att, const int* __restrict__ dst,
    float* __restrict__ mseg, int E) {
  const int e = blockIdx.x * 256 + threadIdx.x;
  if (e < E) atomicMax(&mseg[dst[e]], att[e]);
}

__global__ __launch_bounds__(256) void expsum_kernel(
    const float* __restrict__ att, const int* __restrict__ dst,
    const float* __restrict__ mseg, float* __restrict__ exv,
    float* __restrict__ denom, int E) {
  const int e = blockIdx.x * 256 + threadIdx.x;
  if (e < E) {
    const int dn = dst[e];
    const float x = expf(att[e] - mseg[dn]);
    exv[e] = x;
    atomicAdd(&denom[dn], x);
  }
}

__global__ __launch_bounds__(256) void agg_kernel(
    const float* __restrict__ Vf, const float* __restrict__ exv,
    const float* __restrict__ denom, const int* __restrict__ src,
    const int* __restrict__ dst, float* __restrict__ agg) {
  const int e = blockIdx.x;
  const int t = threadIdx.x;
  const int s = src[e], dn = dst[e];
  const float w = exv[e] / fmaxf(denom[dn], 1e-30f);
  if (t < DF)
    atomicAdd(&agg[(size_t)dn * DF + t], w * Vf[(size_t)s * DF + t]);
}

// ======================= merge + batchnorm + tanh =======================
__global__ __launch_bounds__(256) void combine_kernel(
    const float* __restrict__ Sf, float* __restrict__ agg, size_t total) {
  const size_t i = (size_t)blockIdx.x * 256 + threadIdx.x;
  if (i < total) agg[i] = (Sf[i] + agg[i]) * (1.0f / 3.0f);
}

__global__ __launch_bounds__(256) void stats_kernel(
    const float* __restrict__ np_, float* __restrict__ colsum,
    float* __restrict__ colsq, int nrows) {
  const int d = threadIdx.x;
  if (d >= DF) return;
  const int r0 = blockIdx.x * 128;
  const int rend = min(r0 + 128, nrows);
  float s = 0.f, ss = 0.f;
  for (int r = r0; r < rend; ++r) {
    const float v = np_[(size_t)r * DF + d];
    s += v;
    ss = fmaf(v, v, ss);
  }
  atomicAdd(&colsum[d], s);
  atomicAdd(&colsq[d], ss);
}

__global__ __launch_bounds__(256) void scaleshift_kernel(
    const float* __restrict__ colsum, const float* __restrict__ colsq,
    const float* __restrict__ gamma, const float* __restrict__ beta,
    float* __restrict__ scale, float* __restrict__ shift, int nrows) {
  const int d = threadIdx.x;
  if (d >= DF) return;
  const float inv_n = 1.0f / (float)nrows;
  const float mean = colsum[d] * inv_n;
  const float var = colsq[d] * inv_n - mean * mean;
  const float sc = gamma[d] * rsqrtf(var + 1e-5f);
  scale[d] = sc;
  shift[d] = beta[d] - mean * sc;
}

__global__ __launch_bounds__(256) void final_kernel(
    const float* __restrict__ np_, const float* __restrict__ scale,
    const float* __restrict__ shift, float* __restrict__ out, size_t total) {
  const size_t i = (size_t)blockIdx.x * 256 + threadIdx.x;
  if (i < total) {
    const int d = (int)(i % DF);
    out[i] = tanhf(fmaf(np_[i], scale[d], shift[d]));
  }
}

// ======================= launch =======================
extern "C" void kernel_launch(void* const* d_in, const int* in_sizes, int n_in,
                              void* d_out, int out_size, void* d_ws,
                              size_t ws_size, hipStream_t stream) {
  const float* X = (const float*)d_in[0];       // n_in_feats [N,D]
  const float* rfeats = (const float*)d_in[1];  // r_feats [R,D]
  // d_in[2] norm, d_in[6..9] wO/wI, d_in[25] in_mask: dead w.r.t. outputs
  const float* loop = (const float*)d_in[3];
  const float* relA = (const float*)d_in[4];
  const float* wcomp = (const float*)d_in[5];
  const float* wSw = (const float*)d_in[10];
  const float* wSb = (const float*)d_in[11];
  const float* wRw = (const float*)d_in[12];
  const float* wRb = (const float*)d_in[13];
  const float* kw = (const float*)d_in[14];
  const float* kb = (const float*)d_in[15];
  const float* qw = (const float*)d_in[16];
  const float* qb = (const float*)d_in[17];
  const float* vw = (const float*)d_in[18];
  const float* vb = (const float*)d_in[19];
  const float* gamma = (const float*)d_in[20];
  const float* beta = (const float*)d_in[21];
  const int* src = (const int*)d_in[22];
  const int* dst = (const int*)d_in[23];
  const int* ety = (const int*)d_in[24];

  const int N = in_sizes[0] / DF;
  const int R = in_sizes[1] / DF;
  const int E = in_sizes[22];
  const int B = in_sizes[5] / R;
  const size_t ND = (size_t)N * DF;

  // ---- workspace layout (floats + ints), ~138 MB total ----
  float* ws = (float*)d_ws;
  size_t o = 0;
  float* Mrel = ws + o;   o += (size_t)R * DF * DF;
  float* Kf = ws + o;     o += ND;
  float* Qf = ws + o;     o += ND;
  float* Vf = ws + o;     o += ND;
  float* Sf = ws + o;     o += ND;
  float* agg = ws + o;    o += ND;
  float* att = ws + o;    o += (size_t)E;
  float* exv = ws + o;    o += (size_t)E;
  float* mseg = ws + o;   o += (size_t)N;
  float* denom = ws + o;  o += (size_t)N;
  float* colsum = ws + o; o += 256;
  float* colsq = ws + o;  o += 256;
  float* scale = ws + o;  o += 256;
  float* shift = ws + o;  o += 256;
  float* sbias = ws + o;  o += 256;
  int* cnt = (int*)(ws + o);    o += 256;
  int* offs = (int*)(ws + o);   o += 320;   // R+1 entries
  int* cursor = (int*)(ws + o); o += 256;
  int* eidx = (int*)(ws + o);   o += (size_t)E;

  float* nout = (float*)d_out;
  float* rout = nout + ND;

  // 1. basis-mix relation matrices
  dim3 gM((DF * DF + 255) / 256, R);
  relM_kernel<<<gM, 256, 0, stream>>>(relA, wcomp, Mrel, B);

  // 2. fold self-loop subtraction into the S bias
  sbias_kernel<<<1, 256, 0, stream>>>(loop, wSw, wSb, sbias);

  // 3. fused K/Q/V/S node projections (WMMA, LDS-staged weight panels)
  const int row_tiles = (N + 15) / 16;
  dim3 gN((row_tiles + 7) / 8, CTILES);
  node_proj_wmma<<<gN, 256, 0, stream>>>(X, kw, kb, qw, qb, vw, vb, wSw, sbias,
                                         Kf, Qf, Vf, Sf, N);

  // 4. r_out = r_feats @ wR^T + wR_b
  const int rtiles = ((R + 15) / 16) * CTILES;
  gemm_xwT_wmma<<<(rtiles + 7) / 8, 256, 0, stream>>>(rfeats, wRw, wRb, rout, R);

  // 5. init accumulators + histogram buckets
  init_kernel<<<(int)((ND + 255) / 256), 256, 0, stream>>>(
      agg, mseg, denom, colsum, colsq, cnt, R, N, ND);

  // 6. counting-sort edges by relation type
  hist_kernel<<<(E + 255) / 256, 256, 0, stream>>>(ety, cnt, E);
  scan_kernel<<<1, 256, 0, stream>>>(cnt, offs, cursor, R);
  scatter_kernel<<<(E + 255) / 256, 256, 0, stream>>>(ety, cursor, eidx, E);

  // 7. grouped per-edge bilinear attention via WMMA
  dim3 gA(R, (E + EPB - 1) / EPB);  // oversized in y; inactive chunks exit
  edge_att_wmma<<<gA, 128, 0, stream>>>(Kf, Qf, Mrel, eidx, offs, att);

  // 8-10. segment softmax + weighted aggregation
  attmax_kernel<<<(E + 255) / 256, 256, 0, stream>>>(att, dst, mseg, E);
  expsum_kernel<<<(E + 255) / 256, 256, 0, stream>>>(att, dst, mseg, exv, denom, E);
  agg_kernel<<<E, 256, 0, stream>>>(Vf, exv, denom, src, dst, agg);

  // 11. merge self-loop branch
  combine_kernel<<<(int)((ND + 255) / 256), 256, 0, stream>>>(Sf, agg, ND);

  // 12-14. batchnorm (batch stats) + tanh
  stats_kernel<<<(N + 127) / 128, 256, 0, stream>>>(agg, colsum, colsq, N);
  scaleshift_kernel<<<1, 256, 0, stream>>>(colsum, colsq, gamma, beta, scale,
                                           shift, N);
  final_kernel<<<(int)((ND + 255) / 256), 256, 0, stream>>>(agg, scale, shift,
                                                            nout, ND);
}